// BlockwiseAttention_37374805409933
// MI455X (gfx1250) — compile-verified
//
#include <hip/hip_runtime.h>

typedef __attribute__((ext_vector_type(2))) float v2f;
typedef __attribute__((ext_vector_type(8))) float v8f;

#define BH_TOTAL   64          // B*H = 4*16
#define Q_LEN      512
#define K_LEN      512
#define HEAD_D     64
#define ATTN_SCALE 0.125f      // 64^-0.5

// One wave32 computes one 16x16 tile of S = Q*K^T using V_WMMA_F32_16X16X4_F32.
__global__ __launch_bounds__(256) void attn_scores_wmma_f32(
    const float* __restrict__ query,   // (BH, 512, 64)
    const float* __restrict__ key,     // (BH, 512, 64)
    float* __restrict__ scores,        // (BH, 512, 512)
    const int* __restrict__ qbi,       // scalar query_block_idx
    const int* __restrict__ kbi)       // scalar kv_block_idx
{
    const int qb = qbi[0];
    const int kb = kbi[0];

    const int wave = threadIdx.x >> 5;
    const int lane = threadIdx.x & 31;
    const int half = lane >> 4;        // 0: lanes 0-15, 1: lanes 16-31
    const int l16  = lane & 15;

    const unsigned tile = blockIdx.x * 8u + (unsigned)wave;  // 65536 tiles
    const unsigned bh   = tile >> 10;                        // / (32*32)
    const unsigned rem  = tile & 1023u;
    const int tm = (int)(rem >> 5);                          // tile row   (M/16)
    const int tn = (int)(rem & 31u);                         // tile col   (N/16)

    // A fragment: Q row (tm*16 + l16); per k-step, lane holds K = kk + 2*half, +1
    const float* qrow = query + (size_t)bh * Q_LEN * HEAD_D
                              + (size_t)(tm * 16 + l16) * HEAD_D + 2 * half;
    // B fragment: B[k][n] = Key[n][k]; lane holds Key row (tn*16 + l16), same K slots
    const float* krow = key   + (size_t)bh * K_LEN * HEAD_D
                              + (size_t)(tn * 16 + l16) * HEAD_D + 2 * half;

    v8f acc = {};
#pragma unroll
    for (int kk = 0; kk < HEAD_D; kk += 4) {
        v2f a = *(const v2f*)(qrow + kk);
        v2f b = *(const v2f*)(krow + kk);
        acc = __builtin_amdgcn_wmma_f32_16x16x4_f32(
            /*neg_a=*/false, a, /*neg_b=*/false, b,
            /*c_mod=*/(short)0, acc, /*reuse_a=*/false, /*reuse_b=*/false);
    }

    // C/D layout: VGPR v, lane -> (M = v + 8*half, N = l16) within the tile.
    float* obase = scores + (size_t)bh * Q_LEN * K_LEN;
    const int n = tn * 16 + l16;
    const bool all_mask = (kb > qb);
    const bool causal   = (kb == qb);
    const float neg_inf = -__builtin_inff();
#pragma unroll
    for (int v = 0; v < 8; ++v) {
        const int m = tm * 16 + v + 8 * half;
        const float val = acc[v] * ATTN_SCALE;
        const bool masked = all_mask || (causal && (n > m));
        obase[(size_t)m * K_LEN + n] = masked ? neg_inf : val;
    }
}

__global__ __launch_bounds__(256) void copy_value_f32x4(
    const float4* __restrict__ src, float4* __restrict__ dst, int n4)
{
    const int i = blockIdx.x * blockDim.x + threadIdx.x;
    if (i < n4) dst[i] = src[i];
}

extern "C" void kernel_launch(void* const* d_in, const int* in_sizes, int n_in,
                              void* d_out, int out_size, void* d_ws, size_t ws_size,
                              hipStream_t stream)
{
    const float* query = (const float*)d_in[0];
    const float* key   = (const float*)d_in[1];
    const float* value = (const float*)d_in[2];
    const int*   qbi   = (const int*)d_in[3];
    const int*   kbi   = (const int*)d_in[4];

    float* scores = (float*)d_out;
    const size_t score_elems = (size_t)BH_TOTAL * Q_LEN * K_LEN;   // 16,777,216
    float* value_out = scores + score_elems;
    const int value_elems = BH_TOTAL * K_LEN * HEAD_D;             // 2,097,152

    // 65536 tiles, 8 waves per 256-thread block -> 8192 blocks
    attn_scores_wmma_f32<<<8192, 256, 0, stream>>>(query, key, scores, qbi, kbi);

    const int n4 = value_elems / 4;                                // 524,288 float4
    copy_value_f32x4<<<(n4 + 255) / 256, 256, 0, stream>>>(
        (const float4*)value, (float4*)value_out, n4);
}